// MultiBoxLoss_2138893713901
// MI455X (gfx1250) — compile-verified
//
#include <hip/hip_runtime.h>
#include <stdint.h>

#define Bn 64
#define Pn 25200
#define Gn 48
#define THRESH 0.35f
#define NEGPOS 7
#define CHUNK 10240   // floats staged in LDS per TDM transfer (40 KB)

typedef float v2f __attribute__((ext_vector_type(2)));
typedef float v8f __attribute__((ext_vector_type(8)));
typedef unsigned v4u __attribute__((ext_vector_type(4)));
typedef int v8i __attribute__((ext_vector_type(8)));
typedef int v4i __attribute__((ext_vector_type(4)));

__device__ __forceinline__ float sl1(float x) {
  float d = fabsf(x);
  return d < 1.f ? 0.5f * d * d : d - 0.5f;
}

// Wave32 reduction of TWO independent values with one V_WMMA_F32_16X16X4_F32.
// A (16x4) selects rows 0-7 <- K0,K1 (x-lanes) and rows 8-15 <- K2,K3 (y-lanes);
// B (4x16) carries per-lane {x,y}. D row0 (lanes 0-15) = x[n]+x[n+16],
// D row8 (lanes 16-31) = y[n]+y[n+16]. Butterfly xor 1/2/4/8 stays within each
// 16-lane half, so lane0 -> sum(x), lane16 -> sum(y).
__device__ __forceinline__ float wave_reduce2_wmma(float x, float y) {
  int lane = threadIdx.x & 31;
  int m = lane & 15;
  float a = (((lane >= 16) ? 1 : 0) == ((m >= 8) ? 1 : 0)) ? 1.f : 0.f;
  v2f A = {a, a};
  v2f Bv = {x, y};
  v8f Cc = {0.f, 0.f, 0.f, 0.f, 0.f, 0.f, 0.f, 0.f};
  Cc = __builtin_amdgcn_wmma_f32_16x16x4_f32(false, A, false, Bv, (short)0, Cc,
                                             false, false);
  float v = Cc[0];
  v += __shfl_xor(v, 1);
  v += __shfl_xor(v, 2);
  v += __shfl_xor(v, 4);
  v += __shfl_xor(v, 8);
  return v;  // lane0: sum(x); lane16: sum(y)
}

// Tensor Data Mover: DMA a 1-D run of n 4-byte elements from global into LDS.
// D# group0: count=1, lds_addr, global_addr, type=2. group1: data_size=4B,
// tensor_dim0=n, tensor_dim1=1, tile_dim0=n, tile_dim1=1, dim0_stride=n.
// This toolchain exposes the 6-arg builtin:
//   (uint32x4 g0, int32x8 g1, int32x4 g2, int32x4 g3, int32x8 pad, i32 cpol)
__device__ __forceinline__ void tdm_load_1d(unsigned lds_off, const float* src,
                                            int nelem) {
  unsigned long long ga = (unsigned long long)(uintptr_t)src;
  unsigned nd = (unsigned)nelem;
  v4u g0;
  g0[0] = 1u;                                                // count = 1
  g0[1] = lds_off;                                           // lds_addr
  g0[2] = (unsigned)ga;                                      // global_addr lo
  g0[3] = (unsigned)((ga >> 32) & 0x1FFFFFFull) | (2u << 30);// hi + type=2
  v8i g1;
  g1[0] = (int)(2u << 16);                    // data_size = 2 (4 bytes)
  g1[1] = (int)((nd & 0xFFFFu) << 16);        // tensor_dim0 [15:0]
  g1[2] = (int)(((nd >> 16) & 0xFFFFu) | (1u << 16)); // dim0 hi | tensor_dim1=1
  g1[3] = (int)((nd & 0xFFFFu) << 16);        // tile_dim0
  g1[4] = 1;                                  // tile_dim1 = 1
  g1[5] = (int)nd;                            // tensor_dim0_stride lo
  g1[6] = 0;
  g1[7] = 0;
  v4i z4 = {0, 0, 0, 0};
  v8i z8 = {0, 0, 0, 0, 0, 0, 0, 0};
  __builtin_amdgcn_tensor_load_to_lds(g0, g1, z4, z4, z8, 0);
}

// ----- shared GT staging (normalized truths, landmarks, labels, mask, area)
struct Gt {
  float tr[Gn][4];
  float lm[Gn][10];
  float area[Gn];
  int label[Gn];
  int mask[Gn];
};

__device__ __forceinline__ void load_gt(Gt& gs, int b,
                                        const float* __restrict__ gt_bboxes,
                                        const int* __restrict__ gt_labels,
                                        const float* __restrict__ gt_landmarks,
                                        const int* __restrict__ gt_num,
                                        const int* __restrict__ img_shape) {
  float w = (float)img_shape[b * 2 + 1];
  float h = (float)img_shape[b * 2 + 0];
  int ng = gt_num[b];
  for (int g = threadIdx.x; g < Gn; g += blockDim.x) {
    const float* bb = gt_bboxes + ((size_t)b * Gn + g) * 4;
    float x1 = bb[0] / w, y1 = bb[1] / h, x2 = bb[2] / w, y2 = bb[3] / h;
    gs.tr[g][0] = x1; gs.tr[g][1] = y1; gs.tr[g][2] = x2; gs.tr[g][3] = y2;
    gs.area[g] = (x2 - x1) * (y2 - y1);
    const float* lmp = gt_landmarks + ((size_t)b * Gn + g) * 10;
    for (int k = 0; k < 10; ++k) gs.lm[g][k] = lmp[k] / ((k & 1) ? h : w);
    int lbl = gt_labels[(size_t)b * Gn + g];
    if (lbl == 0) lbl = 1;
    if (gs.lm[g][0] < 0.f) lbl = -1;
    gs.label[g] = lbl;
    gs.mask[g] = (g < ng) ? 1 : 0;
  }
}

// ----- K0: zero workspace accumulators
__global__ void k_init(unsigned long long* bpo, int* num_pos, int* has_valid,
                       float* facc, int* iacc) {
  int i = blockIdx.x * blockDim.x + threadIdx.x;
  if (i < Bn * Gn) bpo[i] = 0ull;
  if (i < Bn) { num_pos[i] = 0; has_valid[i] = 0; }
  if (i < 4) facc[i] = 0.f;
  if (i < 2) iacc[i] = 0;
}

// ----- K1: IoU + best_truth (per prior) + best_prior (per gt, packed argmax)
__global__ __launch_bounds__(256) void k_match(
    const float* __restrict__ priors, const float* __restrict__ gt_bboxes,
    const int* __restrict__ gt_labels, const float* __restrict__ gt_landmarks,
    const int* __restrict__ gt_num, const int* __restrict__ img_shape,
    float* __restrict__ bto, int* __restrict__ bti,
    unsigned long long* __restrict__ bpo) {
  __shared__ Gt gs;
  __shared__ unsigned long long best[Gn];
  int b = blockIdx.y;
  load_gt(gs, b, gt_bboxes, gt_labels, gt_landmarks, gt_num, img_shape);
  for (int g = threadIdx.x; g < Gn; g += blockDim.x) best[g] = 0ull;
  __syncthreads();

  int p = blockIdx.x * blockDim.x + threadIdx.x;
  if (p < Pn) {
    const float4 pr = ((const float4*)priors)[p];
    float px1 = pr.x - pr.z * 0.5f, py1 = pr.y - pr.w * 0.5f;
    float px2 = pr.x + pr.z * 0.5f, py2 = pr.y + pr.w * 0.5f;
    float areaB = pr.z * pr.w;
    float bestv = -2.f;
    int bestg = 0;
    for (int g = 0; g < Gn; ++g) {
      float ix1 = fmaxf(gs.tr[g][0], px1), iy1 = fmaxf(gs.tr[g][1], py1);
      float ix2 = fminf(gs.tr[g][2], px2), iy2 = fminf(gs.tr[g][3], py2);
      float iw = fmaxf(ix2 - ix1, 0.f), ih = fmaxf(iy2 - iy1, 0.f);
      float inter = iw * ih;
      float iou = inter / (gs.area[g] + areaB - inter);
      float v = gs.mask[g] ? iou : -1.f;
      if (v > bestv) { bestv = v; bestg = g; }
      if (gs.mask[g]) {
        // monotone pack: larger IoU wins, then SMALLER prior index (jnp.argmax)
        unsigned long long pk =
            ((unsigned long long)__float_as_uint(iou) << 32) |
            (unsigned long long)(0xFFFFFFFFu - (unsigned)p);
        if (pk > best[g]) atomicMax(&best[g], pk);
      }
    }
    bto[(size_t)b * Pn + p] = bestv;
    bti[(size_t)b * Pn + p] = bestg;
  }
  __syncthreads();
  for (int g = threadIdx.x; g < Gn; g += blockDim.x)
    if (best[g]) atomicMax(&bpo[(size_t)b * Gn + g], best[g]);
}

// ----- K2: sequential fix-up (later j wins), overlap fill = 2.0, has_valid
__global__ void k_fixup(const unsigned long long* __restrict__ bpo,
                        const int* __restrict__ gt_num, float* __restrict__ bto,
                        int* __restrict__ bti, int* __restrict__ has_valid) {
  int b = blockIdx.x * blockDim.x + threadIdx.x;
  if (b >= Bn) return;
  int ng = gt_num[b];
  int hv = 0;
  for (int j = 0; j < Gn; ++j) {
    unsigned long long pk = bpo[(size_t)b * Gn + j];
    float ov = __uint_as_float((unsigned)(pk >> 32));
    unsigned idx = 0xFFFFFFFFu - (unsigned)pk;
    if (j < ng) {
      if (idx < (unsigned)Pn) {
        bti[(size_t)b * Pn + idx] = j;             // later j overwrites
        if (ov >= 0.2f) bto[(size_t)b * Pn + idx] = 2.0f;
      }
      if (ov >= 0.2f) hv = 1;
    }
  }
  has_valid[b] = hv;
}

// ----- K3: encode + smooth-L1 sums + CE + per-batch positive counts
__global__ __launch_bounds__(256) void k_loss(
    const float* __restrict__ loc_data, const float* __restrict__ conf_data,
    const float* __restrict__ landm_data, const float* __restrict__ priors,
    const float* __restrict__ gt_bboxes, const int* __restrict__ gt_labels,
    const float* __restrict__ gt_landmarks, const int* __restrict__ gt_num,
    const int* __restrict__ img_shape, const float* __restrict__ bto,
    const int* __restrict__ bti, const int* __restrict__ has_valid,
    float* __restrict__ lc, int* __restrict__ num_pos, float* __restrict__ facc,
    int* __restrict__ iacc) {
  __shared__ Gt gs;
  __shared__ float accf[3];
  __shared__ int acci[2];
  int b = blockIdx.y;
  load_gt(gs, b, gt_bboxes, gt_labels, gt_landmarks, gt_num, img_shape);
  if (threadIdx.x < 3) accf[threadIdx.x] = 0.f;
  if (threadIdx.x < 2) acci[threadIdx.x] = 0;
  __syncthreads();

  int p = blockIdx.x * blockDim.x + threadIdx.x;
  int hv = has_valid[b];
  float ll = 0.f, lmv = 0.f, cep = 0.f;
  bool pos = false, pos1 = false;
  if (p < Pn) {
    size_t bp = (size_t)b * Pn + p;
    float o = bto[bp];
    int ti = bti[bp];
    int conf = (hv && o >= THRESH) ? gs.label[ti] : 0;
    pos = conf != 0;
    pos1 = conf > 0;
    const float2 cd = ((const float2*)conf_data)[bp];
    float mx = fmaxf(cd.x, cd.y);
    float logz = mx + logf(expf(cd.x - mx) + expf(cd.y - mx));
    float ce = logz - (pos ? cd.y : cd.x);
    lc[bp] = pos ? 0.f : ce;
    if (pos) {
      cep = ce;
      const float4 pr = ((const float4*)priors)[p];
      const float* m = gs.tr[ti];
      float gcx = ((m[0] + m[2]) * 0.5f - pr.x) / (0.1f * pr.z);
      float gcy = ((m[1] + m[3]) * 0.5f - pr.y) / (0.1f * pr.w);
      float gw = logf((m[2] - m[0]) / pr.z) / 0.2f;
      float gh = logf((m[3] - m[1]) / pr.w) / 0.2f;
      const float4 ld = ((const float4*)loc_data)[bp];
      ll = sl1(ld.x - gcx) + sl1(ld.y - gcy) + sl1(ld.z - gw) + sl1(ld.w - gh);
      if (pos1) {
        const float* lmp = landm_data + bp * 10;
        for (int i = 0; i < 5; ++i) {
          float tx = (gs.lm[ti][2 * i] - pr.x) / (0.1f * pr.z);
          float ty = (gs.lm[ti][2 * i + 1] - pr.y) / (0.1f * pr.w);
          lmv += sl1(lmp[2 * i] - tx) + sl1(lmp[2 * i + 1] - ty);
        }
      }
    }
  }
  unsigned long long balp = __ballot(pos);
  unsigned long long balp1 = __ballot(pos1);
  float r0 = wave_reduce2_wmma(ll, lmv);   // lane0 = sum ll, lane16 = sum lmv
  float r1 = wave_reduce2_wmma(cep, 0.f);  // lane0 = sum cep
  int lane = threadIdx.x & 31;
  if (lane == 0) {
    atomicAdd(&accf[0], r0);
    atomicAdd(&accf[2], r1);
    atomicAdd(&acci[0], (int)__popcll(balp));
    atomicAdd(&acci[1], (int)__popcll(balp1));
  }
  if (lane == 16) atomicAdd(&accf[1], r0);
  __syncthreads();
  if (threadIdx.x == 0) {
    atomicAdd(&facc[0], accf[0]);   // loss_l
    atomicAdd(&facc[1], accf[1]);   // loss_landm
    atomicAdd(&facc[2], accf[2]);   // ce over pos
    atomicAdd(&num_pos[b], acci[0]);
    atomicAdd(&iacc[0], acci[0]);   // N
    atomicAdd(&iacc[1], acci[1]);   // N1
  }
}

// ----- K4: hard-negative mining: exact top-K sum via 8-bit radix select.
// CE >= 0 so the float bit pattern is order-isomorphic. Row chunks are staged
// into LDS with the Tensor Data Mover (wave 0 issues, s_wait_tensorcnt).
__global__ __launch_bounds__(1024) void k_mine(const float* __restrict__ lc,
                                               const int* __restrict__ num_pos,
                                               float* __restrict__ facc) {
  __shared__ float stage[CHUNK];
  __shared__ unsigned hist[256];
  __shared__ unsigned s_bin, s_need;
  __shared__ float s_acc[2];
  int b = blockIdx.x;
  const float* row = lc + (size_t)b * Pn;
  long long K = (long long)NEGPOS * num_pos[b];
  if (K > Pn - 1) K = Pn - 1;
  if (K <= 0) return;  // uniform per block; no WMMA executed on this path
  int tid = threadIdx.x;
  if (tid < 2) s_acc[tid] = 0.f;
  unsigned lds_base = (unsigned)(uintptr_t)(&stage[0]);

  unsigned prefix = 0, pmask = 0, need = (unsigned)K;
  for (int shift = 24; shift >= 0; shift -= 8) {
    if (tid < 256) hist[tid] = 0;
    __syncthreads();
    for (int base = 0; base < Pn; base += CHUNK) {
      int n = min(CHUNK, Pn - base);
      if (tid < 32) {  // wave 0 drives the DMA
        tdm_load_1d(lds_base, row + base, n);
        __builtin_amdgcn_s_wait_tensorcnt(0);
      }
      __syncthreads();
      for (int i = tid; i < n; i += 1024) {
        unsigned u = __float_as_uint(stage[i]);
        if ((u & pmask) == prefix) atomicAdd(&hist[(u >> shift) & 255u], 1u);
      }
      __syncthreads();
    }
    if (tid == 0) {
      unsigned nd = need;
      int sel = 0;
      for (int bin = 255; bin >= 0; --bin) {
        unsigned c = hist[bin];
        if (c >= nd) { sel = bin; break; }
        nd -= c;
      }
      s_bin = (unsigned)sel;
      s_need = nd;
    }
    __syncthreads();
    prefix |= s_bin << shift;
    pmask |= 255u << shift;
    need = s_need;
    __syncthreads();
  }
  float T = __uint_as_float(prefix);  // exact K-th largest value

  float psum = 0.f;
  int pcnt = 0;
  for (int base = 0; base < Pn; base += CHUNK) {
    int n = min(CHUNK, Pn - base);
    if (tid < 32) {
      tdm_load_1d(lds_base, row + base, n);
      __builtin_amdgcn_s_wait_tensorcnt(0);
    }
    __syncthreads();
    for (int i = tid; i < n; i += 1024) {
      float v = stage[i];
      if (__float_as_uint(v) > prefix) { psum += v; pcnt++; }
    }
    __syncthreads();
  }
  float r = wave_reduce2_wmma(psum, (float)pcnt);
  int lane = tid & 31;
  if (lane == 0) atomicAdd(&s_acc[0], r);
  if (lane == 16) atomicAdd(&s_acc[1], r);
  __syncthreads();
  if (tid == 0) {
    // ties at T: each contributes exactly T, so index tie-break is irrelevant
    float neg = s_acc[0] + ((float)K - s_acc[1]) * T;
    atomicAdd(&facc[3], neg);
  }
}

// ----- K5: finalize the three scalar losses
__global__ void k_final(const float* __restrict__ facc,
                        const int* __restrict__ iacc, float* __restrict__ out) {
  if (threadIdx.x == 0 && blockIdx.x == 0) {
    float N = fmaxf((float)iacc[0], 1.f);
    float N1 = fmaxf((float)iacc[1], 1.f);
    out[0] = facc[0] / N;
    out[1] = (facc[2] + facc[3]) / N;
    out[2] = facc[1] / N1;
  }
}

extern "C" void kernel_launch(void* const* d_in, const int* in_sizes, int n_in,
                              void* d_out, int out_size, void* d_ws,
                              size_t ws_size, hipStream_t stream) {
  const float* loc = (const float*)d_in[0];
  const float* confd = (const float*)d_in[1];
  const float* landm = (const float*)d_in[2];
  const float* priors = (const float*)d_in[3];
  const float* gtb = (const float*)d_in[4];
  const int* gtl = (const int*)d_in[5];
  const float* gtlm = (const float*)d_in[6];
  const int* gtn = (const int*)d_in[7];
  const int* imgs = (const int*)d_in[8];

  unsigned long long* bpo = (unsigned long long*)d_ws;   // [B*G]
  float* bto = (float*)(bpo + (size_t)Bn * Gn);          // [B*P]
  int* bti = (int*)(bto + (size_t)Bn * Pn);              // [B*P]
  float* lc = (float*)(bti + (size_t)Bn * Pn);           // [B*P]
  int* num_pos = (int*)(lc + (size_t)Bn * Pn);           // [B]
  int* has_valid = num_pos + Bn;                         // [B]
  float* facc = (float*)(has_valid + Bn);                // [4]
  int* iacc = (int*)(facc + 4);                          // [2]

  k_init<<<dim3((Bn * Gn + 255) / 256), 256, 0, stream>>>(bpo, num_pos,
                                                          has_valid, facc, iacc);
  dim3 gridM((Pn + 255) / 256, Bn);
  k_match<<<gridM, 256, 0, stream>>>(priors, gtb, gtl, gtlm, gtn, imgs, bto,
                                     bti, bpo);
  k_fixup<<<1, 64, 0, stream>>>(bpo, gtn, bto, bti, has_valid);
  k_loss<<<gridM, 256, 0, stream>>>(loc, confd, landm, priors, gtb, gtl, gtlm,
                                    gtn, imgs, bto, bti, has_valid, lc, num_pos,
                                    facc, iacc);
  k_mine<<<Bn, 1024, 0, stream>>>(lc, num_pos, facc);
  k_final<<<1, 1, 0, stream>>>(facc, iacc, (float*)d_out);
}